// StateLIFLegacy_23218593202607
// MI455X (gfx1250) — compile-verified
//
#include <hip/hip_runtime.h>

// LIF scan: spikes[t] = H(tau*mem + x[t] - vth), mem resets on spike.
// Pure streaming, memory-bound kernel: 268 MB of traffic, ~11.5us floor
// at 23.3 TB/s. One thread owns one float4 across all T=4 timesteps.

typedef float v4f __attribute__((ext_vector_type(4)));

namespace {
constexpr int   kT    = 4;
constexpr int   kB    = 64;
constexpr int   kC    = 128;
constexpr int   kH    = 32;
constexpr int   kW    = 32;
constexpr int   kCHW  = kC * kH * kW;          // 131072 (per-batch spatial volume)
constexpr long long kN = (long long)kB * kCHW; // 8388608 elements per timestep
constexpr int   kN4   = (int)(kN / 4);         // 2097152 float4 per timestep
constexpr int   kCHW4 = kCHW / 4;              // 32768 (power of two -> mask)
constexpr float kTau  = 0.25f;
constexpr float kVth  = 1.0f;
constexpr int   kBlock = 256;                  // 8 wave32s per workgroup
}  // namespace

__global__ __launch_bounds__(kBlock) void lif_scan_kernel(
    const v4f* __restrict__ x,      // [T, B, C, H, W] viewed as [T, N4] float4
    const v4f* __restrict__ mem0,   // [1, C, H, W]    viewed as [CHW4]  float4
    v4f* __restrict__ out) {        // [T, B, C, H, W] viewed as [T, N4] float4
  const int i = blockIdx.x * kBlock + threadIdx.x;   // [0, N4)

  // Initial membrane state: broadcast over batch. Reused by 64 threads per
  // element -> regular (cacheable) load so L2/WGP$ serve the broadcast.
  v4f m = mem0[i & (kCHW4 - 1)];

#pragma unroll
  for (int t = 0; t < kT; ++t) {
    const size_t idx = (size_t)t * (size_t)kN4 + (size_t)i;
    // x is touched exactly once: non-temporal b128 load (gfx1250 TH=NT),
    // address independent of the recurrence so all 4 loads issue up front.
    v4f xt = __builtin_nontemporal_load(&x[idx]);

    v4f s;
#pragma unroll
    for (int c = 0; c < 4; ++c) {
      const float mc = fmaf(kTau, m[c], xt[c]);  // mem = tau*mem + x[t]
      const bool fire = mc > kVth;               // hard-threshold forward
      s[c] = fire ? 1.0f : 0.0f;                 // spike
      m[c] = fire ? 0.0f : mc;                   // mem *= (1 - spike)
    }

    // Spikes are write-once streaming output: non-temporal b128 store.
    __builtin_nontemporal_store(s, &out[idx]);
  }
}

extern "C" void kernel_launch(void* const* d_in, const int* in_sizes, int n_in,
                              void* d_out, int out_size, void* d_ws, size_t ws_size,
                              hipStream_t stream) {
  (void)in_sizes; (void)n_in; (void)d_ws; (void)ws_size; (void)out_size;
  const v4f* x    = (const v4f*)d_in[0];  // [T*B*C*H*W] fp32
  const v4f* mem0 = (const v4f*)d_in[1];  // [C*H*W] fp32
  v4f*       out  = (v4f*)d_out;          // [T*B*C*H*W] fp32

  const int blocks = kN4 / kBlock;  // 8192 blocks, exact cover, no tail
  lif_scan_kernel<<<blocks, kBlock, 0, stream>>>(x, mem0, out);
}